// CenterLoss_6880537608649
// MI455X (gfx1250) — compile-verified
//
#include <hip/hip_runtime.h>

typedef __attribute__((ext_vector_type(2))) float v2f;
typedef __attribute__((ext_vector_type(8))) float v8f;

#define NW      8            // waves per block
#define TPB     (NW * 32)
#define KSPLIT  4            // K-dimension split across blocks
#define TPB2    256          // final-reduce block size

// Block (g, s): 16-sample group g, K-chunk s. Waves interleave inside the chunk.
// Each wave computes diag(X_g * Cg^T) for its K-slice via V_WMMA_F32_16X16X4_F32
// plus per-row ||x||^2, ||c||^2 partials. Block writes 16 un-clamped row
// partials of (||x||^2 + ||c||^2 - 2 x.c) to ws[(g*KSPLIT+s)*16 + m].
__global__ __launch_bounds__(TPB) void center_loss_partial(
    const float* __restrict__ x, const int* __restrict__ labels,
    const float* __restrict__ centers, float* __restrict__ ws,
    int B, int D)
{
    __shared__ float s_part[NW * 16];

    const int g    = blockIdx.x / KSPLIT;  // 16-sample group
    const int s    = blockIdx.x % KSPLIT;  // K-chunk
    const int lane = threadIdx.x & 31;
    const int wid  = threadIdx.x >> 5;

    const int m    = lane & 15;            // matrix row (== col) served by this lane
    const int row  = g * 16 + m;
    const int rowc = (row < B) ? row : (B - 1);
    const int lab  = labels[rowc];

    const float* __restrict__ xrow = x       + (size_t)rowc * (size_t)D;
    const float* __restrict__ crow = centers + (size_t)lab  * (size_t)D;

    // A 16x4 f32 layout: lanes 0-15 carry K = k0,k0+1 ; lanes 16-31 carry K = k0+2,k0+3.
    const int koff = (lane >> 4) * 2;
    const int step = KSPLIT * NW * 4;

    v8f   acc0 = {}, acc1 = {};            // two accumulators -> split WMMA RAW chain
    float xsq = 0.0f, csq = 0.0f;

    int k0 = (s * NW + wid) * 4;

    // Manually unrolled x4: 8 b64 loads in flight before the first wmma waits.
    for (; k0 + 3 * step < D; k0 += 4 * step) {
        float2 a0 = *(const float2*)(xrow + k0 +            koff);
        float2 b0 = *(const float2*)(crow + k0 +            koff);
        float2 a1 = *(const float2*)(xrow + k0 +     step + koff);
        float2 b1 = *(const float2*)(crow + k0 +     step + koff);
        float2 a2 = *(const float2*)(xrow + k0 + 2 * step + koff);
        float2 b2 = *(const float2*)(crow + k0 + 2 * step + koff);
        float2 a3 = *(const float2*)(xrow + k0 + 3 * step + koff);
        float2 b3 = *(const float2*)(crow + k0 + 3 * step + koff);

        xsq = fmaf(a0.x, a0.x, fmaf(a0.y, a0.y, xsq));
        csq = fmaf(b0.x, b0.x, fmaf(b0.y, b0.y, csq));
        v2f av0 = {a0.x, a0.y}; v2f bv0 = {b0.x, b0.y};
        acc0 = __builtin_amdgcn_wmma_f32_16x16x4_f32(false, av0, false, bv0,
                                                     (short)0, acc0, false, false);
        xsq = fmaf(a1.x, a1.x, fmaf(a1.y, a1.y, xsq));
        csq = fmaf(b1.x, b1.x, fmaf(b1.y, b1.y, csq));
        v2f av1 = {a1.x, a1.y}; v2f bv1 = {b1.x, b1.y};
        acc1 = __builtin_amdgcn_wmma_f32_16x16x4_f32(false, av1, false, bv1,
                                                     (short)0, acc1, false, false);
        xsq = fmaf(a2.x, a2.x, fmaf(a2.y, a2.y, xsq));
        csq = fmaf(b2.x, b2.x, fmaf(b2.y, b2.y, csq));
        v2f av2 = {a2.x, a2.y}; v2f bv2 = {b2.x, b2.y};
        acc0 = __builtin_amdgcn_wmma_f32_16x16x4_f32(false, av2, false, bv2,
                                                     (short)0, acc0, false, false);
        xsq = fmaf(a3.x, a3.x, fmaf(a3.y, a3.y, xsq));
        csq = fmaf(b3.x, b3.x, fmaf(b3.y, b3.y, csq));
        v2f av3 = {a3.x, a3.y}; v2f bv3 = {b3.x, b3.y};
        acc1 = __builtin_amdgcn_wmma_f32_16x16x4_f32(false, av3, false, bv3,
                                                     (short)0, acc1, false, false);
    }
    // Remainder (not taken for D % (4*KSPLIT*NW*4) == 0, e.g. D = 1024).
    for (; k0 < D; k0 += step) {
        float2 a = *(const float2*)(xrow + k0 + koff);
        float2 b = *(const float2*)(crow + k0 + koff);
        xsq = fmaf(a.x, a.x, fmaf(a.y, a.y, xsq));
        csq = fmaf(b.x, b.x, fmaf(b.y, b.y, csq));
        v2f av = {a.x, a.y}; v2f bv = {b.x, b.y};
        acc0 = __builtin_amdgcn_wmma_f32_16x16x4_f32(false, av, false, bv,
                                                     (short)0, acc0, false, false);
    }

    v8f acc = acc0 + acc1;

    // Lanes L and L^16 hold complementary K-phases of the same row m: combine norms.
    xsq += __shfl_xor(xsq, 16);
    csq += __shfl_xor(csq, 16);

    // diag(D): row i (0..7) -> VGPR i, lane i ; row 8+r -> VGPR r, lane 24+r.
    const bool valid = (lane < 8) || (lane >= 24);
    const int  r = lane & 7;
    float diag;
    switch (r) {
        case 0: diag = acc[0]; break;
        case 1: diag = acc[1]; break;
        case 2: diag = acc[2]; break;
        case 3: diag = acc[3]; break;
        case 4: diag = acc[4]; break;
        case 5: diag = acc[5]; break;
        case 6: diag = acc[6]; break;
        default: diag = acc[7]; break;
    }

    if (valid) {
        s_part[wid * 16 + m] = xsq + csq - 2.0f * diag;   // this wave's K-slice partial
    }
    __syncthreads();

    if (threadIdx.x < 16) {
        float d = 0.0f;
        #pragma unroll
        for (int w = 0; w < NW; ++w) d += s_part[w * 16 + threadIdx.x];
        ws[(size_t)blockIdx.x * 16 + threadIdx.x] = d;    // un-clamped partial
    }
}

// Sum K-splits per row, clamp (AFTER masking, like reference), reduce
// deterministically (fixed order; no float atomics), scale and add the
// masked-zero constant term.
__global__ __launch_bounds__(TPB2) void center_loss_final(
    const float* __restrict__ ws, float* __restrict__ out,
    int B, float invB, float baseTerm)
{
    __shared__ float red[TPB2 / 32];

    const int tid  = threadIdx.x;
    const int lane = tid & 31;
    const int wid  = tid >> 5;

    float acc = 0.0f;
    for (int b = tid; b < B; b += TPB2) {
        const int gIdx = b >> 4;
        const int m    = b & 15;
        float d = 0.0f;
        #pragma unroll
        for (int s = 0; s < KSPLIT; ++s)
            d += ws[(size_t)(gIdx * KSPLIT + s) * 16 + m];
        d = fminf(fmaxf(d, 1e-12f), 1e12f);
        acc += d;
    }
    #pragma unroll
    for (int off = 16; off > 0; off >>= 1) acc += __shfl_xor(acc, off);
    if (lane == 0) red[wid] = acc;
    __syncthreads();

    if (tid < 32) {
        float v = (lane < (TPB2 / 32)) ? red[lane] : 0.0f;
        #pragma unroll
        for (int off = 16; off > 0; off >>= 1) v += __shfl_xor(v, off);
        if (tid == 0) out[0] = fmaf(v, invB, baseTerm);
    }
}

extern "C" void kernel_launch(void* const* d_in, const int* in_sizes, int n_in,
                              void* d_out, int out_size, void* d_ws, size_t ws_size,
                              hipStream_t stream)
{
    const float* x       = (const float*)d_in[0];   // [B, D] f32
    const int*   labels  = (const int*)d_in[1];     // [B] int
    const float* centers = (const float*)d_in[2];   // [C, D] f32
    float*       out     = (float*)d_out;           // scalar f32

    const int B = in_sizes[1];
    const int D = in_sizes[0] / B;
    const int C = in_sizes[2] / D;

    const int ngroups = (B + 15) / 16;
    float* ws = (float*)d_ws;                       // ngroups*KSPLIT*16 floats

    center_loss_partial<<<ngroups * KSPLIT, TPB, 0, stream>>>(x, labels, centers, ws, B, D);

    // masked-zero entries each clip to 1e-12: (B*C - B)*1e-12 / B = (C-1)*1e-12
    const float baseTerm = (float)((double)(C - 1) * 1e-12);
    center_loss_final<<<1, TPB2, 0, stream>>>(ws, out, B, 1.0f / (float)B, baseTerm);
}